// NewAttnDecoderRNN_75213467288171
// MI455X (gfx1250) — compile-verified
//
#include <hip/hip_runtime.h>
#include <hip/hip_bf16.h>

// Shapes from the reference
#define V_  50257
#define E_  1024
#define H_  1024
#define L_  512
#define EH_ 2048
#define H3_ 3072

typedef __attribute__((ext_vector_type(2))) float v2f;
typedef __attribute__((ext_vector_type(8))) float v8f;

#define TILE_K 64
#define LDS_STRIDE 68      // 64 + 4 pad -> lane n hits banks 4n..4n+3, conflict-free
#define ROWS_PER_BLOCK 128 // 8 waves * 16 rows

// ---------------------------------------------------------------------------
// Generic WMMA-based GEMV: y[r] = dot(W[r, 0:K], x[0:K]) + b[r]
// W row-major (nrows x K), K multiple of 64.
// Each wave computes 16 rows using V_WMMA_F32_16X16X4_F32 with
//   A = x broadcast over M, B = 16 weight rows over N.
// ---------------------------------------------------------------------------
__global__ __launch_bounds__(256)
void wmma_gemv_kernel(const float* __restrict__ W, const float* __restrict__ x,
                      const float* __restrict__ bias, float* __restrict__ y,
                      int nrows, int K) {
  __shared__ __align__(16) float lds_w[ROWS_PER_BLOCK * LDS_STRIDE];
  __shared__ float lds_x[TILE_K];

  const int tid  = threadIdx.x;
  const int lane = tid & 31;
  const int wave = tid >> 5;
  const int n    = lane & 15;   // output column / weight row within tile
  const int hi   = lane >> 4;   // 0: K=k,k+1   1: K=k+2,k+3
  const int row_base = blockIdx.x * ROWS_PER_BLOCK;

  v8f acc = {};

  for (int k0 = 0; k0 < K; k0 += TILE_K) {
    // Stage x chunk (broadcast operand)
    if (tid < TILE_K) lds_x[tid] = x[k0 + tid];

    // Stage 128x64 weight tile, coalesced float4 loads
#pragma unroll
    for (int i = 0; i < 8; ++i) {
      int idx = i * 256 + tid;        // 0..2047
      int r   = idx >> 4;             // 0..127
      int c4  = (idx & 15) << 2;      // 0,4,...,60
      int gr  = row_base + r;
      if (gr >= nrows) gr = nrows - 1;      // safe clamp, results discarded
      const float4 v = *reinterpret_cast<const float4*>(W + (size_t)gr * K + k0 + c4);
      *reinterpret_cast<float4*>(&lds_w[r * LDS_STRIDE + c4]) = v;
    }
    __syncthreads();

    // Prefetch next k-chunk (global_prefetch_b8): 128 rows x 2 cachelines
    if (k0 + TILE_K < K) {
      int gr = row_base + (tid >> 1);
      if (gr >= nrows) gr = nrows - 1;
      const char* p = (const char*)(W + (size_t)gr * K + (k0 + TILE_K)) + (tid & 1) * 128;
      __builtin_prefetch(p, 0, 1);
    }

    // 16 WMMAs over this chunk, all lanes active (EXEC all-ones)
    const float* wrow = &lds_w[(wave * 16) * LDS_STRIDE + n * LDS_STRIDE];
#pragma unroll
    for (int k = 0; k < TILE_K; k += 4) {
      const int kk = k + hi * 2;
      v2f a; a.x = lds_x[kk];  a.y = lds_x[kk + 1];   // A: x broadcast over M
      v2f b; b.x = wrow[kk];   b.y = wrow[kk + 1];    // B: weight rows over N
      acc = __builtin_amdgcn_wmma_f32_16x16x4_f32(
          /*neg_a=*/false, a, /*neg_b=*/false, b,
          /*c_mod=*/(short)0, acc, /*reuse_a=*/false, /*reuse_b=*/false);
    }
    __syncthreads();
  }

  // D[m][n] identical over m; acc[0] at lane n (n<16) holds row row_base+wave*16+n
  const int r = row_base + wave * 16 + n;
  if (hi == 0 && r < nrows) y[r] = acc[0] + bias[r];
}

// ---------------------------------------------------------------------------
// Setup: embedded = emb[idx]; x1 = [embedded, h]; x2[0:E] = embedded
// ---------------------------------------------------------------------------
__global__ __launch_bounds__(256)
void setup_kernel(const int* __restrict__ input, const float* __restrict__ hidden,
                  const float* __restrict__ emb, float* __restrict__ x1,
                  float* __restrict__ x2) {
  const int idx = input[0];
  const int t = blockIdx.x * 256 + threadIdx.x;      // 0..1023
  const float e = emb[(size_t)idx * E_ + t];
  x1[t] = e;
  x2[t] = e;
  x1[E_ + t] = hidden[t];
}

// ---------------------------------------------------------------------------
// Softmax over 512 logits -> aw (ws) and attn_weights (output)
// ---------------------------------------------------------------------------
__global__ __launch_bounds__(512)
void softmax512_kernel(const float* __restrict__ logits, float* __restrict__ aw,
                       float* __restrict__ aw_out) {
  __shared__ float red[512];
  const int t = threadIdx.x;
  const float v = logits[t];
  red[t] = v; __syncthreads();
  for (int s = 256; s > 0; s >>= 1) {
    if (t < s) red[t] = fmaxf(red[t], red[t + s]);
    __syncthreads();
  }
  const float m = red[0];
  __syncthreads();
  const float e = expf(v - m);
  red[t] = e; __syncthreads();
  for (int s = 256; s > 0; s >>= 1) {
    if (t < s) red[t] += red[t + s];
    __syncthreads();
  }
  const float w = e / red[0];
  aw[t] = w;
  aw_out[t] = w;
}

// ---------------------------------------------------------------------------
// attn_applied (split-K): partial[g][j] = sum_{l in rows g} aw[l]*enc[l][j]
// 64 blocks = 16 row-groups x 4 col-groups
// ---------------------------------------------------------------------------
__global__ __launch_bounds__(256)
void attn_partial_kernel(const float* __restrict__ aw, const float* __restrict__ enc,
                         float* __restrict__ partial) {
  const int bg = blockIdx.x >> 2;                  // 0..15
  const int bc = blockIdx.x & 3;                   // 0..3
  const int j  = bc * 256 + threadIdx.x;           // 0..1023
  const int l0 = bg * 32;
  float s = 0.f;
#pragma unroll 4
  for (int l = 0; l < 32; ++l) s += aw[l0 + l] * enc[(size_t)(l0 + l) * H_ + j];
  partial[bg * H_ + j] = s;
}

__global__ __launch_bounds__(256)
void attn_reduce_kernel(const float* __restrict__ partial, float* __restrict__ x2) {
  const int j = blockIdx.x * 256 + threadIdx.x;
  float s = 0.f;
#pragma unroll
  for (int g = 0; g < 16; ++g) s += partial[g * H_ + j];
  x2[E_ + j] = s;                                   // second half of x2
}

// ---------------------------------------------------------------------------
// GRU gating -> h_new (ws copy + output copy)
// ---------------------------------------------------------------------------
__global__ __launch_bounds__(256)
void gru_kernel(const float* __restrict__ gi, const float* __restrict__ gh,
                const float* __restrict__ h, float* __restrict__ hnew,
                float* __restrict__ hnew_out) {
  const int i = blockIdx.x * 256 + threadIdx.x;     // 0..1023
  const float r = 1.f / (1.f + expf(-(gi[i] + gh[i])));
  const float z = 1.f / (1.f + expf(-(gi[H_ + i] + gh[H_ + i])));
  const float nn = tanhf(gi[2 * H_ + i] + r * gh[2 * H_ + i]);
  const float hv = h[i];
  const float out = (1.f - z) * nn + z * hv;
  hnew[i] = out;
  hnew_out[i] = out;
}

// ---------------------------------------------------------------------------
// log_softmax over N=50257: block-max -> block-sumexp -> write (no atomics)
// ---------------------------------------------------------------------------
#define RED_BLOCKS 64

__global__ __launch_bounds__(256)
void reduce_max_kernel(const float* __restrict__ v, int n, float* __restrict__ bmax) {
  __shared__ float red[256];
  float m = -3.402823466e38f;
  for (int i = blockIdx.x * 256 + threadIdx.x; i < n; i += gridDim.x * 256)
    m = fmaxf(m, v[i]);
  red[threadIdx.x] = m; __syncthreads();
  for (int s = 128; s > 0; s >>= 1) {
    if (threadIdx.x < s) red[threadIdx.x] = fmaxf(red[threadIdx.x], red[threadIdx.x + s]);
    __syncthreads();
  }
  if (threadIdx.x == 0) bmax[blockIdx.x] = red[0];
}

__global__ __launch_bounds__(256)
void reduce_sumexp_kernel(const float* __restrict__ v, int n,
                          const float* __restrict__ bmax, float* __restrict__ bsum) {
  __shared__ float red[256];
  float gm = -3.402823466e38f;
#pragma unroll
  for (int i = 0; i < RED_BLOCKS; ++i) gm = fmaxf(gm, bmax[i]);
  float s = 0.f;
  for (int i = blockIdx.x * 256 + threadIdx.x; i < n; i += gridDim.x * 256)
    s += expf(v[i] - gm);
  red[threadIdx.x] = s; __syncthreads();
  for (int st = 128; st > 0; st >>= 1) {
    if (threadIdx.x < st) red[threadIdx.x] += red[threadIdx.x + st];
    __syncthreads();
  }
  if (threadIdx.x == 0) bsum[blockIdx.x] = red[0];
}

__global__ __launch_bounds__(256)
void logsoftmax_write_kernel(const float* __restrict__ v, int n,
                             const float* __restrict__ bmax,
                             const float* __restrict__ bsum,
                             float* __restrict__ out) {
  float gm = -3.402823466e38f;
  float gs = 0.f;
#pragma unroll
  for (int i = 0; i < RED_BLOCKS; ++i) { gm = fmaxf(gm, bmax[i]); gs += bsum[i]; }
  const float c = gm + logf(gs);
  for (int i = blockIdx.x * 256 + threadIdx.x; i < n; i += gridDim.x * 256)
    out[i] = v[i] - c;
}

// ---------------------------------------------------------------------------
// Launch: 12 kernels on `stream`, graph-capture safe.
// ---------------------------------------------------------------------------
extern "C" void kernel_launch(void* const* d_in, const int* in_sizes, int n_in,
                              void* d_out, int out_size, void* d_ws, size_t ws_size,
                              hipStream_t stream) {
  const int*   input  = (const int*)  d_in[0];
  const float* hidden = (const float*)d_in[1];
  const float* enc    = (const float*)d_in[2];
  const float* emb    = (const float*)d_in[3];
  const float* attn_W = (const float*)d_in[4];
  const float* attn_b = (const float*)d_in[5];
  const float* W_ih   = (const float*)d_in[6];
  const float* W_hh   = (const float*)d_in[7];
  const float* b_ih   = (const float*)d_in[8];
  const float* b_hh   = (const float*)d_in[9];
  const float* out_W  = (const float*)d_in[10];
  const float* out_b  = (const float*)d_in[11];

  float* out = (float*)d_out;
  float* out_logprobs = out;              // [V_]
  float* out_hnew     = out + V_;         // [H_]
  float* out_attn     = out + V_ + H_;    // [L_]

  float* ws = (float*)d_ws;
  float* x1      = ws;                    // [2048]  = [embedded, h]
  float* x2      = ws + 2048;             // [2048]  = [embedded, attn_applied]
  float* logitsA = ws + 4096;             // [512]
  float* aw      = ws + 4608;             // [512]
  float* partial = ws + 5120;             // [16*1024]
  float* gi      = ws + 21504;            // [3072]
  float* gh      = ws + 24576;            // [3072]
  float* hnew    = ws + 27648;            // [1024]
  float* outlog  = ws + 28672;            // [50257]
  float* bmax    = ws + 78929;            // [64]
  float* bsum    = ws + 78993;            // [64]

  // 1. gather + concat
  setup_kernel<<<4, 256, 0, stream>>>(input, hidden, emb, x1, x2);
  // 2. attention logits: 512 x 2048 GEMV (WMMA)
  wmma_gemv_kernel<<<L_ / ROWS_PER_BLOCK, 256, 0, stream>>>(attn_W, x1, attn_b, logitsA, L_, EH_);
  // 3. softmax -> attn_weights
  softmax512_kernel<<<1, 512, 0, stream>>>(logitsA, aw, out_attn);
  // 4. attn_applied = aw @ enc (split-K + combine)
  attn_partial_kernel<<<64, 256, 0, stream>>>(aw, enc, partial);
  attn_reduce_kernel<<<4, 256, 0, stream>>>(partial, x2);
  // 5. gi = W_ih @ x2 + b_ih   (3072 x 2048, WMMA)
  wmma_gemv_kernel<<<H3_ / ROWS_PER_BLOCK, 256, 0, stream>>>(W_ih, x2, b_ih, gi, H3_, EH_);
  // 6. gh = W_hh @ h + b_hh    (3072 x 1024, WMMA)
  wmma_gemv_kernel<<<H3_ / ROWS_PER_BLOCK, 256, 0, stream>>>(W_hh, x1 + E_, b_hh, gh, H3_, H_);
  // 7. GRU gate -> h_new
  gru_kernel<<<4, 256, 0, stream>>>(gi, gh, hidden, hnew, out_hnew);
  // 8. out logits: 50257 x 1024 GEMV (WMMA, dominant: ~201 MB)
  wmma_gemv_kernel<<<(V_ + ROWS_PER_BLOCK - 1) / ROWS_PER_BLOCK, 256, 0, stream>>>(
      out_W, hnew, out_b, outlog, V_, H_);
  // 9. log_softmax
  reduce_max_kernel<<<RED_BLOCKS, 256, 0, stream>>>(outlog, V_, bmax);
  reduce_sumexp_kernel<<<RED_BLOCKS, 256, 0, stream>>>(outlog, V_, bmax, bsum);
  logsoftmax_write_kernel<<<RED_BLOCKS, 256, 0, stream>>>(outlog, V_, bmax, bsum, out_logprobs);
}